// NonlinearQuantizer_12360915878147
// MI455X (gfx1250) — compile-verified
//
#include <hip/hip_runtime.h>
#include <cfloat>
#include <cstdint>

// ---------------------------------------------------------------------------
// Block-wise absmax nonlinear quantizer (quant-dequant) for MI455X / gfx1250.
// Memory-bound streamer: input staged global->LDS via the Tensor Data Mover
// (tensor_load_to_lds, TENSORcnt double-buffering), output written with
// non-temporal b128 stores. No matmul => no WMMA; the CDNA5-specific path
// here is TDM + s_wait_tensorcnt.
// ---------------------------------------------------------------------------

typedef __attribute__((ext_vector_type(4))) float        f32x4;
typedef __attribute__((ext_vector_type(4))) unsigned int u32x4;
typedef __attribute__((ext_vector_type(8))) int          i32x8;
typedef __attribute__((ext_vector_type(4))) int          i32x4;

#define TILE   4096              // floats per TDM tile (16 KB)
#define PASSES (TILE / 1024)     // 256 threads * 4 floats per pass
#define EPSQ   1e-5f

// Build a 1-D TDM descriptor and issue tensor_load_to_lds (clang-23 6-arg form:
// g0 u32x4, g1 i32x8, g2 i32x4, g3 i32x4, extra i32x8, cpol).
// D# Group0: [1:0]=count=1, word1=lds_addr, word2/3=global_addr, type=2.
// D# Group1: data_size=2 (4B), tensor_dim0=n, tensor_dim1=1, tile_dim0=n,
//            tile_dim1=1, tensor_dim0_stride=n, everything else 0.
__device__ __forceinline__ void tdm_stage(const float* gsrc, unsigned lds_off,
                                          unsigned n /* elements, <65536 */) {
  unsigned long long ga = (unsigned long long)(uintptr_t)gsrc;
  u32x4 g0;
  g0[0] = 1u;                                   // count=1, user descriptor
  g0[1] = lds_off;                              // LDS byte address
  g0[2] = (unsigned)ga;                         // global_addr[31:0]
  g0[3] = (unsigned)(ga >> 32) | (2u << 30);    // global_addr[56:32] | type=2
  i32x8 g1;
  g1[0] = (int)(2u << 16);                      // data_size=4B, no multicast
  g1[1] = (int)((n & 0xFFFFu) << 16);           // tensor_dim0[15:0]
  g1[2] = (int)(((n >> 16) & 0xFFFFu) | (1u << 16)); // dim0[31:16] | tensor_dim1=1
  g1[3] = (int)((n & 0xFFFFu) << 16);           // tile_dim0 = n
  g1[4] = 1;                                    // tile_dim1=1, tile_dim2=0
  g1[5] = (int)n;                               // tensor_dim0_stride[31:0]
  g1[6] = 0;
  g1[7] = 0;
  i32x4 gz4 = {0, 0, 0, 0};                     // groups 2/3 unused (<=2D tensor)
  i32x8 gz8 = {0, 0, 0, 0, 0, 0, 0, 0};         // extra group (6-arg form)
  __builtin_amdgcn_tensor_load_to_lds(g0, g1, gz4, gz4, gz8, 0);
}

__global__ void __launch_bounds__(256)
nlq_kernel(const float* __restrict__ w, const float* __restrict__ cent,
           float* __restrict__ out, long long total) {
  __shared__ float sh_c[256];                      // centroids
  __shared__ float sh_b[256];                      // midpoint boundaries (+INF pad)
  __shared__ __align__(16) float buf[2][TILE];     // double-buffered tile

  const int tid = threadIdx.x;

  // Codebook + boundary table into LDS (once per workgroup).
  {
    float c = cent[tid];
    sh_c[tid] = c;
    sh_b[tid] = (tid < 255) ? 0.5f * (c + cent[tid + 1]) : FLT_MAX;
  }

  const long long nTiles = (total + TILE - 1) / TILE;
  long long t = (long long)blockIdx.x;
  int cur = 0;

  // Prime the pipeline: wave 0 issues the TDM DMA for the first tile.
  if (t < nTiles && tid < 32) {
    long long start = t * TILE;
    long long rem = total - start;
    unsigned n = (unsigned)((rem < TILE) ? rem : TILE);
    tdm_stage(w + start, (unsigned)(uintptr_t)(void*)&buf[0][0], n);
  }

  for (; t < nTiles; t += gridDim.x) {
    const long long nt = t + gridDim.x;
    const bool more = nt < nTiles;

    if (tid < 32) {                       // wave 0 drives the DMA pipeline
      if (more) {
        long long ns = nt * TILE;
        long long rem = total - ns;
        unsigned nn = (unsigned)((rem < TILE) ? rem : TILE);
        tdm_stage(w + ns, (unsigned)(uintptr_t)(void*)&buf[cur ^ 1][0], nn);
        __builtin_amdgcn_s_wait_tensorcnt(1);   // older DMA (current tile) done
      } else {
        __builtin_amdgcn_s_wait_tensorcnt(0);   // last tile done
      }
    }
    __syncthreads();                      // tile visible to all 8 waves

    const long long start = t * TILE;
    const long long remll = total - start;
    const int n = (int)((remll < TILE) ? remll : TILE);   // multiple of 64
    const float* lbuf = &buf[cur][0];

#pragma unroll
    for (int p = 0; p < PASSES; ++p) {
      const int li = p * 1024 + tid * 4;  // element offset inside tile
      if (li < n) {
        f32x4 v = *(const f32x4*)(lbuf + li);   // ds_load_b128

        // 64-element block = 16 consecutive lanes * float4. Absmax reduce.
        float m = fmaxf(fmaxf(__builtin_fabsf(v[0]), __builtin_fabsf(v[1])),
                        fmaxf(__builtin_fabsf(v[2]), __builtin_fabsf(v[3])));
        m = fmaxf(m, __shfl_xor(m, 1));
        m = fmaxf(m, __shfl_xor(m, 2));
        m = fmaxf(m, __shfl_xor(m, 4));
        m = fmaxf(m, __shfl_xor(m, 8));
        const float s = fmaxf(m, EPSQ);

        f32x4 o;
#pragma unroll
        for (int k = 0; k < 4; ++k) {
          const float wt = v[k] / s;
          // searchsorted(left) over 255 boundaries: branchless 8-step
          // lower-bound (sh_b[255] = +INF pad keeps idx in [0,255]).
          int idx = 0;
#pragma unroll
          for (int st = 128; st > 0; st >>= 1)
            if (sh_b[idx + st - 1] < wt) idx += st;
          o[k] = sh_c[idx] * s;
        }
        // Streaming output: NT hint, don't pollute the 192 MB L2.
        __builtin_nontemporal_store(o, (f32x4*)(out + start + li));
      }
    }
    __syncthreads();                      // all waves done with buf[cur]
    cur ^= 1;
  }
}

extern "C" void kernel_launch(void* const* d_in, const int* in_sizes, int n_in,
                              void* d_out, int out_size, void* d_ws, size_t ws_size,
                              hipStream_t stream) {
  const float* w    = (const float*)d_in[0];   // (4096, 11008) fp32
  const float* cent = (const float*)d_in[1];   // (256,) fp32, sorted
  float* out = (float*)d_out;

  const long long total  = (long long)in_sizes[0];          // 45,088,768
  const long long nTiles = (total + TILE - 1) / TILE;
  long long g = nTiles < 2048 ? nTiles : 2048;              // persistent-ish grid
  if (g < 1) g = 1;

  nlq_kernel<<<(unsigned)g, 256, 0, stream>>>(w, cent, out, total);
}